// LightweightConv_26396869001405
// MI455X (gfx1250) — compile-verified
//
#include <hip/hip_runtime.h>
#include <math.h>

// Problem constants (match reference)
#define BB   16
#define TT   4000
#define CC   512
#define HH   8
#define KK   31
#define PADL 15

// Tiling
#define TM    64                     // output timesteps per block
#define SROWS 96                     // TM + 2*PADL + 2 slab rows staged in LDS
#define PITCH 98                     // even pitch -> 8B-aligned (row,row+1) pairs, conflict-aware
#define NG    (CC / 128)             // channel groups of 128 contiguous channels
#define NTILE ((TT + TM - 1) / TM)

typedef float v2f __attribute__((ext_vector_type(2)));
typedef float v8f __attribute__((ext_vector_type(8)));

__global__ __launch_bounds__(256)
void lconv_wmma_kernel(const float* __restrict__ x,
                       const float* __restrict__ wraw,
                       const float* __restrict__ braw,
                       float* __restrict__ out)
{
    // xs: head-major transposed slab, column-major in time:
    //   xs[(16*(c%8) + c/8) * PITCH + r] = x[b, blockT-15+r, 128*g + c]
    __shared__ float xs[128 * PITCH];
    __shared__ float wsm[HH * 32];

    const int tid    = threadIdx.x;
    const int blockT = TM * blockIdx.x;
    const int g      = blockIdx.y;
    const int b      = blockIdx.z;

    // ---- per-head softmax of the 31 taps (tiny; threads 0..7) ----
    if (tid < HH) {
        const float* wh = wraw + tid * KK;
        float mx = wh[0];
        #pragma unroll
        for (int k = 1; k < KK; ++k) mx = fmaxf(mx, wh[k]);
        float e[KK];
        float sum = 0.f;
        #pragma unroll
        for (int k = 0; k < KK; ++k) { e[k] = __expf(wh[k] - mx); sum += e[k]; }
        const float inv = 1.f / sum;
        #pragma unroll
        for (int k = 0; k < KK; ++k) wsm[tid * 32 + k] = e[k] * inv;
    }

    // ---- stage 1: global -> LDS with head-major transpose ----
    {
        const float4* x4 = reinterpret_cast<const float4*>(x);
        #pragma unroll
        for (int it = 0; it < (SROWS * 32) / 256; ++it) {
            const int linear = tid + 256 * it;     // 0 .. SROWS*32-1
            const int q = linear & 31;             // float4 slot within 128-ch row
            const int r = linear >> 5;             // slab row 0..95
            const int s = blockT - PADL + r;       // absolute timestep
            float4 v = make_float4(0.f, 0.f, 0.f, 0.f);
            if (s >= 0 && s < TT)
                v = x4[(size_t)(b * TT + s) * (CC / 4) + (size_t)g * 32 + q];
            const float vv[4] = {v.x, v.y, v.z, v.w};
            const int cc0 = q * 4;
            #pragma unroll
            for (int j = 0; j < 4; ++j) {
                const int cc  = cc0 + j;                       // 0..127 within group
                const int col = ((cc & 7) << 4) + (cc >> 3);   // 16*head + chan/8
                xs[col * PITCH + r] = vv[j];
            }
        }
    }
    __syncthreads();

    // ---- stage 2: per-head Toeplitz-band matmul via WMMA f32 16x16x4 ----
    const int w  = tid >> 5;          // wave id == head
    const int l  = tid & 31;
    const int m  = l & 15;            // A row / output row (low part)
    const int kb = (l >> 4) << 1;     // K base: 0 for lanes 0-15, 2 for lanes 16-31
    const int h  = w;

    // A operand (Toeplitz): a[step] = { w[4*step+kb-m], w[4*step+kb-m+1] } (0 outside band)
    v2f a[12];
    #pragma unroll
    for (int step = 0; step < 12; ++step) {
        const int tx = 4 * step + kb - m;
        const int ty = tx + 1;
        a[step].x = (tx >= 0 && tx < KK) ? wsm[h * 32 + tx] : 0.f;
        a[step].y = (ty >= 0 && ty < KK) ? wsm[h * 32 + ty] : 0.f;
    }

    const float bias = braw[h];
    const int   n    = m;                              // B/D column = lane & 15
    const int   c    = g * 128 + n * 8 + h;            // global channel
    const float* bcol = &xs[(h * 16 + n) * PITCH];

    for (int tt2 = 0; tt2 < TM / 16; ++tt2) {
        const int t0 = blockT + 16 * tt2;
        if (t0 >= TT) break;                           // uniform per wave
        v8f acc = {0.f, 0.f, 0.f, 0.f, 0.f, 0.f, 0.f, 0.f};
        #pragma unroll
        for (int step = 0; step < 12; ++step) {
            const int r = 16 * tt2 + 4 * step + kb;    // even -> 8B aligned pair
            const v2f bv = *reinterpret_cast<const v2f*>(bcol + r);
            acc = __builtin_amdgcn_wmma_f32_16x16x4_f32(
                      false, a[step], false, bv, (short)0, acc, false, false);
        }
        // D layout: VGPR v -> row (v + 8*(lane>=16)), N = lane & 15
        const int tbase = t0 + ((l >> 4) << 3);
        #pragma unroll
        for (int v = 0; v < 8; ++v) {
            const int t = tbase + v;
            if (t < TT)
                out[(size_t)(b * TT + t) * CC + c] = acc[v] + bias;
        }
    }
}

extern "C" void kernel_launch(void* const* d_in, const int* in_sizes, int n_in,
                              void* d_out, int out_size, void* d_ws, size_t ws_size,
                              hipStream_t stream) {
    const float* x    = (const float*)d_in[0];   // (B, T, C) fp32
    const float* wgt  = (const float*)d_in[1];   // (H, 1, K) fp32
    const float* bias = (const float*)d_in[2];   // (H,) fp32
    float* out        = (float*)d_out;           // (B, T, C) fp32

    dim3 grid(NTILE, NG, BB);                    // 63 x 4 x 16 blocks
    lconv_wmma_kernel<<<grid, 256, 0, stream>>>(x, wgt, bias, out);
}